// GTLModule_15083925144430
// MI455X (gfx1250) — compile-verified
//
#include <hip/hip_runtime.h>
#include <hip/hip_bf16.h>
#include <math.h>

// Problem constants (match reference)
#define B_    8
#define N_    16384
#define K_    16
#define G_    4
#define CLPE_ 16
#define BN_EPS_ 1e-5f

typedef __attribute__((ext_vector_type(2))) float v2f;
typedef __attribute__((ext_vector_type(8))) float v8f;

// XOR butterfly exchange within groups of 32 lanes via ds_swizzle_b32
// (group-of-32 mode: offset = xor_mask<<10 | and_mask 0x1f). Masks < 16 keep
// the exchange inside each 16-lane half, matching softmax over K=16 neighbors.
template <int MASK>
__device__ __forceinline__ float swz_xor(float v) {
    return __int_as_float(
        __builtin_amdgcn_ds_swizzle(__float_as_int(v), (MASK << 10) | 0x1f));
}
__device__ __forceinline__ float red_sum16(float x) {
    x += swz_xor<1>(x); x += swz_xor<2>(x);
    x += swz_xor<4>(x); x += swz_xor<8>(x);
    return x;
}
__device__ __forceinline__ float red_max16(float x) {
    x = fmaxf(x, swz_xor<1>(x)); x = fmaxf(x, swz_xor<2>(x));
    x = fmaxf(x, swz_xor<4>(x)); x = fmaxf(x, swz_xor<8>(x));
    return x;
}

// One wave (32 lanes) handles one point (b, n).
//   lane = half*16 + j ; j = neighbor index (0..15).
// WMMA f32 16x16x4 (x3, K padded 10->12) computes y = W(16x10) * relf(10x16)
// for all 16 LPE channels x 16 neighbors of this point, fully in fp32.
__global__ __launch_bounds__(256) void gtl_kernel(
    const float* __restrict__ xyz,     // (B, N, 3)
    const float* __restrict__ query,   // (B, 64, N)
    const float* __restrict__ value,   // (B, 64, N, 1)
    const int*   __restrict__ nidx,    // (B, N, K)
    const float* __restrict__ lpeW,    // (16, 10)
    const float* __restrict__ lpeB,    // (16,)
    const float* __restrict__ gam,     // (16,)
    const float* __restrict__ bet,     // (16,)
    const float* __restrict__ bmean,   // (16,)
    const float* __restrict__ bvar,    // (16,)
    float* __restrict__ outp,          // (B, 128, N, 1)
    float* __restrict__ fenc,          // (B, G, 16, N, K)
    float* __restrict__ cent)          // (B, G, N)  (pre-zeroed)
{
    const int lane = threadIdx.x & 31;
    const int wid  = threadIdx.x >> 5;
    const int j    = lane & 15;
    const int half = lane >> 4;

    const int point = blockIdx.x * 8 + wid;   // grid sized so all waves full
    const int b = point >> 14;                // / N_
    const int n = point & (N_ - 1);           // % N_

    // ---- BN scale/shift for this lane's 8 channels (c = r + 8*half) ----
    float sc[8], sh[8];
#pragma unroll
    for (int r = 0; r < 8; ++r) {
        const int c = r + 8 * half;
        const float s = gam[c] * rsqrtf(bvar[c] + BN_EPS_);
        sc[r] = s;
        sh[r] = (lpeB[c] - bmean[c]) * s + bet[c];
    }

    // ---- A operand: lpe_W (16x10 row-major), padded to 16x12, 3 K=4 chunks
    // f32 16x4 A layout: lanes 0-15 -> M=0..15 hold K=0 (v.x), K=1 (v.y);
    //                    lanes 16-31 -> same M hold K=2 (v.x), K=3 (v.y).
    v2f a0, a1, a2;
    {
        const float* wr = lpeW + j * 10;
        const int c0 = 2 * half;
        a0.x = wr[c0];         a0.y = wr[c0 + 1];
        a1.x = wr[4 + c0];     a1.y = wr[4 + c0 + 1];
        a2.x = (half == 0) ? wr[8] : 0.0f;   // ch 10,11 are zero padding
        a2.y = (half == 0) ? wr[9] : 0.0f;
    }

    // ---- gather neighbor, build the 10 relative-position features ----
    const float cx = xyz[((size_t)b * N_ + n) * 3 + 0];
    const float cy = xyz[((size_t)b * N_ + n) * 3 + 1];
    const float cz = xyz[((size_t)b * N_ + n) * 3 + 2];

    // neigh_idx is streamed exactly once -> non-temporal load
    const int idx =
        __builtin_nontemporal_load(&nidx[((size_t)b * N_ + n) * K_ + j]);

    const float nx = xyz[((size_t)b * N_ + idx) * 3 + 0];
    const float ny = xyz[((size_t)b * N_ + idx) * 3 + 1];
    const float nz = xyz[((size_t)b * N_ + idx) * 3 + 2];

    const float rx = cx - nx, ry = cy - ny, rz = cz - nz;
    const float dis = sqrtf(rx * rx + ry * ry + rz * rz);

    float rf[10] = { dis, rx, ry, rz, cx, cy, cz, nx, ny, nz };

    // ---- B operand: relf (10x16, K padded to 12), mirrored 4x16 layout:
    // lanes 0-15 hold K=kb,kb+1 (v.x,v.y); lanes 16-31 hold K=kb+2,kb+3.
    v2f b0, b1, b2;
    {
        const int c0 = 2 * half;
        b0.x = rf[c0];         b0.y = rf[c0 + 1];
        b1.x = rf[4 + c0];     b1.y = rf[4 + c0 + 1];
        b2.x = (half == 0) ? rf[8] : 0.0f;
        b2.y = (half == 0) ? rf[9] : 0.0f;
    }

    // ---- chained fp32 WMMA: D(16x16) = W * relf  (K = 0..3, 4..7, 8..11)
    v8f acc = {};
    acc = __builtin_amdgcn_wmma_f32_16x16x4_f32(false, a0, false, b0,
                                                (short)0, acc, false, false);
    acc = __builtin_amdgcn_wmma_f32_16x16x4_f32(false, a1, false, b1,
                                                (short)0, acc, false, false);
    acc = __builtin_amdgcn_wmma_f32_16x16x4_f32(false, a2, false, b2,
                                                (short)0, acc, false, false);

    // ---- BN + ReLU; D layout: VGPR r, lane -> channel r+8*half, col j ----
    float y[8];
#pragma unroll
    for (int r = 0; r < 8; ++r) {
        const float v = acc[r] * sc[r] + sh[r];
        y[r] = v > 0.0f ? v : 0.0f;
    }

    // ---- f_enc[b][g][c][n][k]: identical across g, coalesced over k=j.
    // 537 MB of write-once output: non-temporal stores keep L2 for gathers.
#pragma unroll
    for (int g = 0; g < G_; ++g) {
        const size_t base =
            (((size_t)(b * G_ + g) * CLPE_ + 8 * half) * N_ + n) * K_ + j;
#pragma unroll
        for (int r = 0; r < 8; ++r)
            __builtin_nontemporal_store(y[r],
                &fenc[base + (size_t)r * ((size_t)N_ * K_)]);
    }

    // ---- attention logits: half-wave owns groups g0=2*half, g1=2*half+1 --
    const int g0 = 2 * half, g1 = 2 * half + 1;
    const float* qb = query + (size_t)b * 64 * N_;
    float lg0 = 0.0f, lg1 = 0.0f;
#pragma unroll
    for (int c = 0; c < 16; ++c) {
        const float q0 = qb[(size_t)(g0 * 16 + c) * N_ + n];
        const float k0 = qb[(size_t)(g0 * 16 + c) * N_ + idx];
        lg0 = fmaf(q0, k0, lg0);
        const float q1 = qb[(size_t)(g1 * 16 + c) * N_ + n];
        const float k1 = qb[(size_t)(g1 * 16 + c) * N_ + idx];
        lg1 = fmaf(q1, k1, lg1);
    }

    // ---- softmax over the 16 neighbors (ds_swizzle XOR butterflies) ----
    const float m0 = red_max16(lg0);
    const float m1 = red_max16(lg1);
    const float e0 = __expf(lg0 - m0);
    const float e1 = __expf(lg1 - m1);
    const float s0 = red_sum16(e0);
    const float s1 = red_sum16(e1);
    const float la0 = e0 / s0;
    const float la1 = e1 / s1;

    // ---- centrality scatter-add: cent[b][g][idx] += la ----
    atomicAdd(&cent[(size_t)(b * G_ + g0) * N_ + idx], la0);
    atomicAdd(&cent[(size_t)(b * G_ + g1) * N_ + idx], la1);

    // ---- out, value half: out[b][g*32+c][n] = sum_j la * value[g*16+c][idx]
    const float* vb = value + (size_t)b * 64 * N_;
#pragma unroll
    for (int c = 0; c < 16; ++c) {
        float t0 = red_sum16(la0 * vb[(size_t)(g0 * 16 + c) * N_ + idx]);
        float t1 = red_sum16(la1 * vb[(size_t)(g1 * 16 + c) * N_ + idx]);
        if (j == c) {
            __builtin_nontemporal_store(t0,
                &outp[((size_t)b * 128 + g0 * 32 + c) * N_ + n]);
            __builtin_nontemporal_store(t1,
                &outp[((size_t)b * 128 + g1 * 32 + c) * N_ + n]);
        }
    }

    // ---- out, f_enc half: need la[g][j] in every lane for own j ----
    float laAll[4];
    laAll[0] = __shfl(la0, j, 32);        // g=0 lives in half 0
    laAll[1] = __shfl(la1, j, 32);        // g=1 lives in half 0
    laAll[2] = __shfl(la0, 16 + j, 32);   // g=2 lives in half 1
    laAll[3] = __shfl(la1, 16 + j, 32);   // g=3 lives in half 1
#pragma unroll
    for (int g = 0; g < G_; ++g) {
#pragma unroll
        for (int r = 0; r < 8; ++r) {
            const float t = red_sum16(laAll[g] * y[r]);
            if (j == r)   // channel c = r + 8*half: both halves write 8 each
                __builtin_nontemporal_store(t,
                    &outp[((size_t)b * 128 + g * 32 + 16 + r + 8 * half) * N_ + n]);
        }
    }
}

extern "C" void kernel_launch(void* const* d_in, const int* in_sizes, int n_in,
                              void* d_out, int out_size, void* d_ws, size_t ws_size,
                              hipStream_t stream) {
    const float* xyz   = (const float*)d_in[0];
    const float* query = (const float*)d_in[1];
    const float* value = (const float*)d_in[2];
    const int*   nidx  = (const int*)d_in[3];
    // d_in[4] = idx_base (unused; batch offset recomputed internally)
    const float* lpeW  = (const float*)d_in[5];
    const float* lpeB  = (const float*)d_in[6];
    const float* gam   = (const float*)d_in[7];
    const float* bet   = (const float*)d_in[8];
    const float* bmean = (const float*)d_in[9];
    const float* bvar  = (const float*)d_in[10];

    float* outp = (float*)d_out;                                   // B*128*N
    float* fenc = outp + (size_t)B_ * 128 * N_;                    // B*G*16*N*K
    float* cent = fenc + (size_t)B_ * G_ * CLPE_ * N_ * K_;        // B*G*N

    // centrality is accumulated with atomics -> must start at zero each call
    hipMemsetAsync(cent, 0, (size_t)B_ * G_ * N_ * sizeof(float), stream);

    const int points = B_ * N_;          // 131072, one wave per point
    dim3 block(256);                     // 8 waves per block
    dim3 grid(points / 8);
    hipLaunchKernelGGL(gtl_kernel, grid, block, 0, stream,
                       xyz, query, value, nidx, lpeW, lpeB,
                       gam, bet, bmean, bvar, outp, fenc, cent);
}